// GATv2Operator_75874892251804
// MI455X (gfx1250) — compile-verified
//
#include <hip/hip_runtime.h>
#include <hip/hip_bf16.h>
#include <math.h>

typedef __attribute__((ext_vector_type(2))) float v2f;
typedef __attribute__((ext_vector_type(8))) float v8f;

#define B_ 4
#define L_ 512
#define E_ 256
#define H_ 8
#define A_ 32
#define LEAK 0.1f

// ---------------------------------------------------------------------------
// Kernel 0: pack the boolean mask [B,1,L,L] (1 byte/elem) into bit-words
// [B][L][L/32] so the attention kernel can test a register bit per (i,j).
// ---------------------------------------------------------------------------
__global__ void gat_pack_mask(const unsigned char* __restrict__ mask,
                              unsigned int* __restrict__ mp, int nwords) {
  int g = blockIdx.x * blockDim.x + threadIdx.x;
  if (g >= nwords) return;
  const unsigned char* src = mask + (size_t)g * 32;
  unsigned int w = 0u;
#pragma unroll
  for (int b = 0; b < 32; ++b) w |= (src[b] ? 1u : 0u) << b;
  mp[g] = w;
}

// ---------------------------------------------------------------------------
// fp32 WMMA GEMM: Y[M,N] = X[M,K] * W[N,K]^T   (one wave per 16x16 tile,
// K-loop of V_WMMA_F32_16X16X4_F32; exact fp32, matches the reference).
// A-layout (ISA 7.12.2, 32-bit A 16x4): lanes 0-15 hold M, VGPR0/1 = K0/K1,
// lanes 16-31 = K2/K3.  B mirrors it (N across lanes).  C/D: VGPR g holds
// row M = g (lanes 0-15) / M = g+8 (lanes 16-31), N = lane&15.
// ---------------------------------------------------------------------------
__global__ void gat_gemm_xwt(const float* __restrict__ X,
                             const float* __restrict__ W,
                             float* __restrict__ Y,
                             int M, int N, int K) {
  const int lane = threadIdx.x & 31;
  const int wave = blockIdx.x * (blockDim.x >> 5) + (threadIdx.x >> 5);
  const int tiles_n = N >> 4;
  const int mt = wave / tiles_n;
  const int nt = wave - mt * tiles_n;
  if (mt * 16 >= M) return;
  const int lm = lane & 15;
  const int hi = lane >> 4;

  const float* xrow = X + (size_t)(mt * 16 + lm) * K + 2 * hi;
  const float* wrow = W + (size_t)(nt * 16 + lm) * K + 2 * hi;

  v8f acc = {};
#pragma unroll 8
  for (int k = 0; k < K; k += 4) {
    v2f a = *(const v2f*)(xrow + k);   // K = k + 2*hi .. +1  (8B aligned)
    v2f b = *(const v2f*)(wrow + k);
    acc = __builtin_amdgcn_wmma_f32_16x16x4_f32(false, a, false, b,
                                                (short)0, acc, false, false);
  }

  float* yrow = Y + (size_t)(mt * 16) * N + nt * 16 + lm;
#pragma unroll
  for (int g = 0; g < 8; ++g) yrow[(size_t)(g + 8 * hi) * N] = acc[g];
}

// ---------------------------------------------------------------------------
// Kernel 2: scores + masked online softmax + attn*k, fused per (b,h).
// Block = 256 threads (8 wave32), one thread per query row i.
// q/k tiles of the whole (b,h) staged in LDS (2 x 64KB); rank-1 part of the
// lrelu split precomputed; inner loop = add + fma(|x|) per element.
//   lrelu(x) = 0.55 x + 0.45 |x|   (slope 0.1, exact)
//   score[i,j] = 0.55*(w.q_j + w.k_i) + sum_a 0.45*w_a*|q[j,a]+k[i,a]|
// ---------------------------------------------------------------------------
__global__ void gat_attn(const float* __restrict__ qk,
                         const float* __restrict__ w_attn,
                         const unsigned int* __restrict__ mp,
                         float* __restrict__ V) {
  const int bh = blockIdx.x >> 1;   // (b,h) pair
  const int half = blockIdx.x & 1;  // which 256 query rows
  const int b = bh >> 3;
  const int h = bh & 7;
  const int tid = threadIdx.x;      // 0..255

  extern __shared__ float lds[];
  float* q_s = lds;                       // [L][A]
  float* k_s = q_s + L_ * A_;             // [L][A]
  float* wq_s = k_s + L_ * A_;            // [L]
  float* wk_s = wq_s + L_;                // [L]
  unsigned int* m_s = (unsigned int*)(wk_s + L_);  // [256][16] mask words

  // ---- stage q,k rows of this (b,h):  qk row = [.. h*64 + (q:0..31 | k:32..63) ..]
  const float* base = qk + ((size_t)b * L_) * (2 * E_) + h * (2 * A_);
  for (int c = tid; c < L_ * 16; c += 256) {       // 16 float4 per row
    const int row = c >> 4;
    const int idx = c & 15;
    const float4 v = *(const float4*)(base + (size_t)row * (2 * E_) + idx * 4);
    float* dst = (idx < 8) ? (q_s + row * A_ + idx * 4)
                           : (k_s + row * A_ + (idx - 8) * 4);
    *(float4*)dst = v;
  }
  // ---- stage packed mask rows for this half
  const unsigned int* mbase = mp + ((size_t)b * L_ + half * 256) * 16;
  for (int c = tid; c < 256 * 16; c += 256) m_s[c] = mbase[c];

  // ---- attention weights (uniform across the grid -> scalar loads)
  float wv[A_], w045[A_];
#pragma unroll
  for (int a = 0; a < A_; ++a) {
    wv[a] = w_attn[a];
    w045[a] = 0.5f * (1.0f - LEAK) * wv[a];
  }
  __syncthreads();

  // ---- rank-1 terms  wq[j] = w.q_j,  wk[i] = w.k_i
  for (int r = tid; r < L_; r += 256) {
    float sq = 0.f, sk = 0.f;
#pragma unroll
    for (int a = 0; a < A_; ++a) {
      sq = fmaf(wv[a], q_s[r * A_ + a], sq);
      sk = fmaf(wv[a], k_s[r * A_ + a], sk);
    }
    wq_s[r] = sq;
    wk_s[r] = sk;
  }
  __syncthreads();

  // ---- per-thread query row
  const int i = half * 256 + tid;
  float ki[A_];
#pragma unroll
  for (int c = 0; c < 8; ++c) {
    float4 v = *(const float4*)(k_s + i * A_ + 4 * c);
    ki[4 * c + 0] = v.x; ki[4 * c + 1] = v.y;
    ki[4 * c + 2] = v.z; ki[4 * c + 3] = v.w;
  }
  const float wk_i = wk_s[i];
  const float cmean = 0.5f * (1.0f + LEAK);

  float m = -__builtin_inff();
  float ssum = 0.f;
  float vals[A_];
#pragma unroll
  for (int a = 0; a < A_; ++a) vals[a] = 0.f;

  for (int jw = 0; jw < 16; ++jw) {
    const unsigned int mw = m_s[tid * 16 + jw];
#pragma unroll 1
    for (int jb = 0; jb < 32; ++jb) {
      if ((mw >> jb) & 1u) continue;          // masked out: skip all work
      const int j = jw * 32 + jb;
      const float* qrow = q_s + j * A_;       // LDS broadcast reads
      const float* krow = k_s + j * A_;
      float kj[A_];
      float part = 0.f;
#pragma unroll
      for (int c = 0; c < 8; ++c) {
        float4 a4 = *(const float4*)(qrow + 4 * c);
        float4 b4 = *(const float4*)(krow + 4 * c);
        kj[4 * c + 0] = b4.x; kj[4 * c + 1] = b4.y;
        kj[4 * c + 2] = b4.z; kj[4 * c + 3] = b4.w;
        part = fmaf(w045[4 * c + 0], fabsf(a4.x + ki[4 * c + 0]), part);
        part = fmaf(w045[4 * c + 1], fabsf(a4.y + ki[4 * c + 1]), part);
        part = fmaf(w045[4 * c + 2], fabsf(a4.z + ki[4 * c + 2]), part);
        part = fmaf(w045[4 * c + 3], fabsf(a4.w + ki[4 * c + 3]), part);
      }
      const float score = fmaf(cmean, wq_s[j] + wk_i, part);

      if (score > m) {                        // rare rescale branch
        const float r = __expf(m - score);
        ssum = fmaf(ssum, r, 1.0f);
#pragma unroll
        for (int a = 0; a < A_; ++a) vals[a] = fmaf(vals[a], r, kj[a]);
        m = score;
      } else {                                // common accumulate branch
        const float p = __expf(score - m);
        ssum += p;
#pragma unroll
        for (int a = 0; a < A_; ++a) vals[a] = fmaf(p, kj[a], vals[a]);
      }
    }
  }

  const float inv = 1.0f / ssum;
  float* vout = V + ((size_t)(b * L_ + i)) * E_ + h * A_;  // [B,L,E] layout
#pragma unroll
  for (int c = 0; c < 8; ++c) {
    float4 o;
    o.x = vals[4 * c + 0] * inv; o.y = vals[4 * c + 1] * inv;
    o.z = vals[4 * c + 2] * inv; o.w = vals[4 * c + 3] * inv;
    *(float4*)(vout + 4 * c) = o;
  }
}

// ---------------------------------------------------------------------------
extern "C" void kernel_launch(void* const* d_in, const int* in_sizes, int n_in,
                              void* d_out, int out_size, void* d_ws, size_t ws_size,
                              hipStream_t stream) {
  const float* emb   = (const float*)d_in[0];          // [B,L,E]
  const unsigned char* mask = (const unsigned char*)d_in[1]; // [B,1,L,L] bool
  const float* W_qk  = (const float*)d_in[2];          // [2E,E]
  const float* wattn = (const float*)d_in[3];          // [A]
  const float* W_out = (const float*)d_in[4];          // [E,E]
  float* out = (float*)d_out;                          // [B,L,E]

  float* qkbuf = (float*)d_ws;                         // [2048][512]  4 MB
  float* Vbuf  = qkbuf + (size_t)(B_ * L_) * (2 * E_); // [2048][256]  2 MB
  unsigned int* mp = (unsigned int*)(Vbuf + (size_t)(B_ * L_) * E_); // 128 KB

  // 0) pack mask bits
  const int nwords = B_ * L_ * (L_ / 32);
  gat_pack_mask<<<(nwords + 255) / 256, 256, 0, stream>>>(mask, mp, nwords);

  // 1) QK projection: [2048,256] x [512,256]^T -> [2048,512]
  //    (2048/16)*(512/16) = 4096 tiles, 8 waves/block
  gat_gemm_xwt<<<4096 / 8, 256, 0, stream>>>(emb, W_qk, qkbuf,
                                             B_ * L_, 2 * E_, E_);

  // 2) fused scores + masked online softmax + attn*k
  const size_t lds_bytes = (size_t)(2 * L_ * A_ + 2 * L_ + 256 * 16) * 4; // ~148 KB
  gat_attn<<<B_ * H_ * 2, 256, lds_bytes, stream>>>(qkbuf, wattn, mp, Vbuf);

  // 3) output projection: [2048,256] x [256,256]^T -> [2048,256]
  gat_gemm_xwt<<<2048 / 8, 256, 0, stream>>>(Vbuf, W_out, out,
                                             B_ * L_, E_, E_);
}